// GCN_82660940579212
// MI455X (gfx1250) — compile-verified
//
#include <hip/hip_runtime.h>
#include <hip/hip_bf16.h>

// ---------------------------------------------------------------------------
// GCN forward:  out = A_sp @ (relu(A_sp @ (x@W1) + b1) @ W2) + b2
// Dense GEMMs use CDNA5 V_WMMA_F32_16X16X4_F32 (f32-exact, matches reference).
// SpMMs are edge-parallel gathers + global f32 atomics (L2-resident support).
// ---------------------------------------------------------------------------

typedef __attribute__((ext_vector_type(2))) float v2f;
typedef __attribute__((ext_vector_type(8))) float v8f;

// C[M,N] = A[M,K] @ B[K,N]; K % 4 == 0, N % 16 == 0.
// Block = 128 threads (4 waves). Wave w -> N-tile w*16. blockIdx.x -> M-tile.
__global__ __launch_bounds__(128) void gemm_wmma_f32(
    const float* __restrict__ A, const float* __restrict__ B,
    float* __restrict__ C, int M, int K, int N) {
  const int wave = threadIdx.x >> 5;
  const int lane = threadIdx.x & 31;
  const int half = lane >> 4;   // 0: K pair {k,k+1} | 1: {k+2,k+3}
  const int l    = lane & 15;   // row (A) / col (B,C) within tile

  const int mBase = blockIdx.x * 16;
  const int nBase = wave * 16;

  // Clamp A row so loads stay in-bounds without perturbing EXEC (WMMA needs
  // all lanes active). Stores are predicated per-row instead.
  int mRow = mBase + l;
  if (mRow >= M) mRow = M - 1;
  const float* __restrict__ Arow = A + (long long)mRow * K;
  const float* __restrict__ Bcol = B + nBase + l;

  v8f acc = {0.f, 0.f, 0.f, 0.f, 0.f, 0.f, 0.f, 0.f};

  for (int k = 0; k < K; k += 4) {
    const int k0 = k + half * 2;
    v2f a;
    a.x = Arow[k0];
    a.y = Arow[k0 + 1];
    v2f b;
    b.x = Bcol[(long long)k0 * N];
    b.y = Bcol[(long long)(k0 + 1) * N];
    // 8 args: (neg_a, A, neg_b, B, c_mod, C, reuse_a, reuse_b)
    acc = __builtin_amdgcn_wmma_f32_16x16x4_f32(
        false, a, false, b, (short)0, acc, false, false);
  }

  // C/D layout: VGPR v, lanes 0-15 -> M=v, N=lane; lanes 16-31 -> M=8+v.
  const int mOut = mBase + half * 8;
  float* __restrict__ Cp = C + (long long)mOut * N + nBase + l;
#pragma unroll
  for (int v = 0; v < 8; ++v) {
    if (mOut + v < M) Cp[(long long)v * N] = acc[v];
  }
}

// out[n*64 + d] = bias[d]   (pre-loads bias so SpMM atomics accumulate on top)
__global__ void init_bias_kernel(float* __restrict__ out,
                                 const float* __restrict__ bias,
                                 long long total) {
  long long t = (long long)blockIdx.x * blockDim.x + threadIdx.x;
  if (t < total) out[t] = bias[t & 63];
}

// out[row[e]] += val[e] * dense[col[e]]  over D=64 columns.
// 16 threads per edge, float4 per thread -> one wave gathers 2 full rows
// as coalesced 256B bursts from the L2-resident support matrix.
__global__ __launch_bounds__(256) void spmm_kernel(
    const int* __restrict__ row, const int* __restrict__ col,
    const float* __restrict__ val, const float* __restrict__ dense,
    float* __restrict__ out, int E) {
  long long t = (long long)blockIdx.x * blockDim.x + threadIdx.x;
  int e = (int)(t >> 4);
  if (e >= E) return;
  const int d = (int)(t & 15) << 2;

  const int r = row[e];
  const int c = col[e];
  const float v = val[e];

  const float4 s = *(const float4*)(dense + (long long)c * 64 + d);
  float* o = out + (long long)r * 64 + d;
  atomicAdd(o + 0, v * s.x);
  atomicAdd(o + 1, v * s.y);
  atomicAdd(o + 2, v * s.z);
  atomicAdd(o + 3, v * s.w);
}

__global__ void relu_kernel(float* __restrict__ x, long long total) {
  long long t = (long long)blockIdx.x * blockDim.x + threadIdx.x;
  if (t < total) x[t] = fmaxf(x[t], 0.0f);
}

extern "C" void kernel_launch(void* const* d_in, const int* in_sizes, int n_in,
                              void* d_out, int out_size, void* d_ws, size_t ws_size,
                              hipStream_t stream) {
  const float* x    = (const float*)d_in[0];   // [N,128]
  const int*   erow = (const int*)  d_in[1];   // [E]
  const int*   ecol = (const int*)  d_in[2];   // [E]
  const float* eval = (const float*)d_in[3];   // [E]
  const float* W1   = (const float*)d_in[4];   // [128,64]
  const float* b1   = (const float*)d_in[5];   // [64]
  const float* W2   = (const float*)d_in[6];   // [64,64]
  const float* b2   = (const float*)d_in[7];   // [64]
  float* out = (float*)d_out;                  // [N,64]

  const int IN_DIM = 128, D = 64;
  const int N = in_sizes[0] / IN_DIM;          // 50000
  const int E = in_sizes[1];                   // 800000
  const long long tot = (long long)N * D;

  float* buf0 = (float*)d_ws;                  // support matrix  [N,64]
  float* buf1 = buf0 + (size_t)N * D;          // hidden h        [N,64]

  const int gemmBlocks = (N + 15) / 16;
  const int elwBlocks  = (int)((tot + 255) / 256);
  const int spmmBlocks = (int)(((long long)E * 16 + 255) / 256);

  // Layer 1: support = x @ W1
  gemm_wmma_f32<<<gemmBlocks, 128, 0, stream>>>(x, W1, buf0, N, IN_DIM, D);
  // h = b1 (broadcast); h += A_sp @ support; h = relu(h)
  init_bias_kernel<<<elwBlocks, 256, 0, stream>>>(buf1, b1, tot);
  spmm_kernel<<<spmmBlocks, 256, 0, stream>>>(erow, ecol, eval, buf0, buf1, E);
  relu_kernel<<<elwBlocks, 256, 0, stream>>>(buf1, tot);

  // Layer 2: support = h @ W2 (reuse buf0)
  gemm_wmma_f32<<<gemmBlocks, 128, 0, stream>>>(buf1, W2, buf0, N, D, D);
  // out = b2 (broadcast); out += A_sp @ support
  init_bias_kernel<<<elwBlocks, 256, 0, stream>>>(out, b2, tot);
  spmm_kernel<<<spmmBlocks, 256, 0, stream>>>(erow, ecol, eval, buf0, out, E);
}